// MultiHeadAttention_67044439491211
// MI455X (gfx1250) — compile-verified
//
#include <hip/hip_runtime.h>

// ---------------------------------------------------------------------------
// MultiHeadAttention (buggy-einsum reference) for MI455X / gfx1250.
//
// Math: softmax row-sums are exactly 1, and the reference einsum
// 'bqkh,bvha->bqha' sums k and v independently, so
//     out[b,q,:] = ((sum_c context[b,c,:]) @ Wkv[:,D:2D]) @ Wout
// identical for every q. query/Wq/mask do not affect the result.
//
// Pipeline (all HBM-bound except the tiny WMMA GEMMs):
//   1) zero_ws           : clear padded 16x512 accumulators
//   2) ctx_colsum        : csum[b,d] = sum_c context[b,c,d]   (16 MB read)
//   3) gemm16_wmma x2    : (16x512)@(512x512) via V_WMMA_F32_16X16X4_F32
//   4) broadcast_out     : out[b,q,:] = row[b,:]              (16 MB write)
// ---------------------------------------------------------------------------

#define B_DIM 4
#define QL    2048
#define CL    2048
#define D_DIM 512

typedef __attribute__((ext_vector_type(2))) float v2f;
typedef __attribute__((ext_vector_type(8))) float v8f;

// ws layout (floats):
//   [    0,  8192)  csumPad : 16 x 512 (rows 0..3 = per-batch context column sums)
//   [ 8192, 16384)  tmpPad  : 16 x 512 (rows 0..3 = csum @ Wkv_v)
//   [16384, 18432)  rowOut  :  4 x 512 (final per-batch output row)

__global__ void zero_ws_kernel(float* __restrict__ ws) {
  int i = blockIdx.x * blockDim.x + threadIdx.x;   // 64 blocks * 256 = 16384
  ws[i] = 0.0f;
}

// Per-batch column sum of context. 512 blocks x 128 threads; each thread owns
// 4 consecutive d (float4), reduces 16 rows, then 4 global f32 atomics.
__global__ void ctx_colsum_kernel(const float* __restrict__ ctx,
                                  float* __restrict__ csum) {
  const int NCHUNK = 128;            // chunks per batch
  const int ROWS   = CL / NCHUNK;    // 16 rows per chunk
  int b     = blockIdx.x / NCHUNK;
  int chunk = blockIdx.x % NCHUNK;
  int d     = threadIdx.x * 4;

  const float* base = ctx + ((size_t)b * CL + (size_t)chunk * ROWS) * D_DIM + d;

  float4 acc = make_float4(0.f, 0.f, 0.f, 0.f);
#pragma unroll 4
  for (int r = 0; r < ROWS; ++r) {
    if (r + 8 < ROWS)
      __builtin_prefetch(base + (size_t)(r + 8) * D_DIM, 0, 1);
    const float4 v = *reinterpret_cast<const float4*>(base + (size_t)r * D_DIM);
    acc.x += v.x; acc.y += v.y; acc.z += v.z; acc.w += v.w;
  }
  float* dst = csum + b * D_DIM + d;
  atomicAdd(dst + 0, acc.x);
  atomicAdd(dst + 1, acc.y);
  atomicAdd(dst + 2, acc.z);
  atomicAdd(dst + 3, acc.w);
}

// C16 = A(16x512, rows>=4 zero) @ Bm[:, colbase : colbase+512].
// One wave per block; block nt computes columns [nt*16, nt*16+16).
// WMMA f32 16x16x4:
//   A 16x4 (2 VGPRs): lane l<16 -> M=l, v0=K0 v1=K1; lanes 16..31 -> K2/K3.
//   B 4x16 (2 VGPRs): mirrored (v0 = K0/K2 rows, v1 = K1/K3 rows, N=lane&15).
//   C/D 16x16 (8 VGPRs): VGPR r, lanes 0..15 -> M=r, lanes 16..31 -> M=r+8.
__global__ void gemm16_wmma_kernel(const float* __restrict__ A,
                                   const float* __restrict__ Bm,
                                   int ldb, int colbase,
                                   float* __restrict__ Cout /* 4x512 */) {
  const int lane = threadIdx.x;      // 0..31
  const int l    = lane & 15;
  const int hs   = lane >> 4;        // half-wave select (K pair 0/1 vs 2/3)
  const int ccol = blockIdx.x * 16 + l;          // output column
  const int bcol0 = colbase + blockIdx.x * 16 + l;

  const float* arow = A + (size_t)l * D_DIM + hs * 2;   // 8B-aligned
  const float* bcol = Bm + bcol0;

  v8f acc = {};
#pragma unroll 4
  for (int k = 0; k < D_DIM; k += 4) {
    v2f a = *reinterpret_cast<const v2f*>(arow + k);    // A[M=l, k+2hs .. +1]
    v2f b;
    b.x = bcol[(size_t)(k + hs * 2 + 0) * ldb];         // B[k+2hs,   N=l]
    b.y = bcol[(size_t)(k + hs * 2 + 1) * ldb];         // B[k+2hs+1, N=l]
    acc = __builtin_amdgcn_wmma_f32_16x16x4_f32(
        /*neg_a=*/false, a, /*neg_b=*/false, b,
        /*c_mod=*/(short)0, acc, /*reuse_a=*/false, /*reuse_b=*/false);
  }

  // Only batch rows M=0..3 are meaningful (lanes 0..15 hold M=r).
  if (hs == 0) {
#pragma unroll
    for (int r = 0; r < 4; ++r)
      Cout[r * D_DIM + ccol] = acc[r];
  }
}

// out[b,q,:] = row[b,:] for all q. One float4 per thread; row (8 KB) stays hot.
__global__ void broadcast_out_kernel(const float* __restrict__ row,
                                     float* __restrict__ out) {
  size_t gid = (size_t)blockIdx.x * blockDim.x + threadIdx.x;  // float4 index
  int b  = (int)(gid >> 18);     // QL*D/4 = 262144 float4 per batch
  int d4 = (int)(gid & 127);     // 512/4 = 128 float4 per row
  const float4 v = reinterpret_cast<const float4*>(row)[b * 128 + d4];
  reinterpret_cast<float4*>(out)[gid] = v;
}

extern "C" void kernel_launch(void* const* d_in, const int* in_sizes, int n_in,
                              void* d_out, int out_size, void* d_ws, size_t ws_size,
                              hipStream_t stream) {
  // setup_inputs order: query(0), context(1), mask(2), Wq(3), Wkv(4), Wout(5).
  // query / mask / Wq are mathematically unused (softmax rows sum to 1).
  const float* context = (const float*)d_in[1];
  const float* Wkv     = (const float*)d_in[4];
  const float* Wout    = (const float*)d_in[5];
  float* out = (float*)d_out;

  float* ws    = (float*)d_ws;
  float* csum  = ws;                   // 16x512 padded
  float* tmp   = ws + 16 * D_DIM;      // 16x512 padded
  float* row   = ws + 32 * D_DIM;      //  4x512

  // 1) zero padded accumulators (rows 4..15 must be zero for the WMMA A tiles)
  zero_ws_kernel<<<64, 256, 0, stream>>>(ws);

  // 2) csum[b,:] = sum_c context[b,c,:]
  ctx_colsum_kernel<<<B_DIM * 128, 128, 0, stream>>>(context, csum);

  // 3) tmp = csumPad @ Wkv[:, D:2D]   (B has row stride 2D, column offset D)
  gemm16_wmma_kernel<<<32, 32, 0, stream>>>(csum, Wkv, 2 * D_DIM, D_DIM, tmp);
  //    row = tmpPad @ Wout
  gemm16_wmma_kernel<<<32, 32, 0, stream>>>(tmp, Wout, D_DIM, 0, row);

  // 4) broadcast to all q positions
  broadcast_out_kernel<<<(B_DIM * QL * D_DIM / 4) / 256, 256, 0, stream>>>(row, out);
}